// AttnReadout_72129680769632
// MI455X (gfx1250) — compile-verified
//
#include <hip/hip_runtime.h>
#include <cstdint>
#include <cstddef>

typedef __attribute__((ext_vector_type(2))) float v2f;
typedef __attribute__((ext_vector_type(8))) float v8f;

#define Bz 32
#define Oz 2
#define Nz 4096
#define Dz 128
#define Mz (Oz * Nz)       // 8192
#define PAIRS (Bz * Oz)    // 64
#define WU_PITCH 132       // LDS row pitch in floats: bank = (4*row + col) % 64 -> conflict-free b64 reads

// workspace layout (in floats)
constexpr size_t OFF_XV    = 0;                                  // PAIRS*Dz      = 8192
constexpr size_t OFF_E     = OFF_XV + (size_t)PAIRS * Dz;        // PAIRS*Mz      = 524288
constexpr size_t OFF_STATS = OFF_E + (size_t)PAIRS * Mz;         // PAIRS*2
constexpr size_t OFF_PART  = OFF_STATS + (size_t)PAIRS * 2;      // PAIRS*32*Dz   = 262144

// ---------------- xv[b,i,:] = Wv[i] @ x[b, i, last_nodes[b,i], :] ----------------
__global__ __launch_bounds__(Dz) void k_xv(const float* __restrict__ x,
                                           const float* __restrict__ Wv,
                                           const int* __restrict__ last,
                                           float* __restrict__ xv) {
  const int p = blockIdx.x;          // pair = b*Oz + i
  const int i = p % Oz;
  const int e = threadIdx.x;         // 0..127
  __shared__ float xl[Dz];
  const int node = last[p];
  const float* xrow = x + ((size_t)p * Nz + node) * Dz;
  xl[e] = xrow[e];
  __syncthreads();
  const float* wrow = Wv + ((size_t)i * Dz + e) * Dz;
  float acc = 0.f;
#pragma unroll 8
  for (int d = 0; d < Dz; ++d) acc += wrow[d] * xl[d];
  xv[(size_t)p * Dz + e] = acc;
}

// ---------------- e[p, m] = We[i] . sigmoid(Wu[i] @ x[b, m] + bu[i] + xv[p]) ----------------
// grid: (8 chunks of 1024 rows, 64 pairs); block: 256 threads = 8 waves; each wave does a 16x128 strip.
__global__ __launch_bounds__(256) void k_scores(const float* __restrict__ x,
                                                const float* __restrict__ Wu,
                                                const float* __restrict__ bu,
                                                const float* __restrict__ We,
                                                const float* __restrict__ xv,
                                                float* __restrict__ e_out) {
  __shared__ float lds_wu[Dz * WU_PITCH];   // Wu[i] row-major [e][d], padded pitch
  const int p = blockIdx.y;
  const int i = p % Oz;
  const int b = p / Oz;
  const int tid = threadIdx.x;

  // cooperative load of Wu[i] (16384 floats) into padded LDS, float4 chunks
  {
    const float4* src = (const float4*)(Wu + (size_t)i * Dz * Dz);
    for (int q = tid; q < (Dz * Dz) / 4; q += 256) {
      const int row = q >> 5;            // Dz/4 = 32 float4 per row
      const int c4  = q & 31;
      *(float4*)(&lds_wu[row * WU_PITCH + c4 * 4]) = src[q];
    }
  }
  __syncthreads();

  const int wave = tid >> 5;
  const int lane = tid & 31;
  const int half = lane >> 4;   // 0: K = k,k+1 ; 1: K = k+2,k+3
  const int l16  = lane & 15;

  // per-lane bias (bu + xv) and We for the 8 E-tiles (N = l16 within each tile)
  float bias_r[8], we_r[8];
#pragma unroll
  for (int t = 0; t < 8; ++t) {
    const int e = t * 16 + l16;
    bias_r[t] = bu[i * Dz + e] + xv[(size_t)p * Dz + e];
    we_r[t]   = We[i * Dz + e];
  }

  const int chunkBase = blockIdx.x * 1024;
  for (int iter = 0; iter < 8; ++iter) {
    const int rowBase = chunkBase + iter * 128 + wave * 16;
    // A operand source: row (l16) of the strip, columns k + 2*half
    const float* arow = x + ((size_t)b * Mz + rowBase + l16) * Dz + 2 * half;

    v8f acc[8];
    const v8f vzero = {0.f, 0.f, 0.f, 0.f, 0.f, 0.f, 0.f, 0.f};
#pragma unroll
    for (int t = 0; t < 8; ++t) acc[t] = vzero;

    for (int k = 0; k < Dz; k += 4) {
      const v2f a = *(const v2f*)(arow + k);
#pragma unroll
      for (int t = 0; t < 8; ++t) {
        const v2f bvec = *(const v2f*)(&lds_wu[(t * 16 + l16) * WU_PITCH + k + 2 * half]);
        acc[t] = __builtin_amdgcn_wmma_f32_16x16x4_f32(
            /*neg_a=*/false, a, /*neg_b=*/false, bvec,
            /*c_mod=*/(short)0, acc[t], /*reuse_a=*/false, /*reuse_b=*/false);
      }
    }

    // epilogue: sigmoid + dot with We, accumulated per output row j (M = j + 8*half)
    float partial[8];
#pragma unroll
    for (int j = 0; j < 8; ++j) partial[j] = 0.f;
#pragma unroll
    for (int t = 0; t < 8; ++t) {
#pragma unroll
      for (int j = 0; j < 8; ++j) {
        const float u = acc[t][j] + bias_r[t];
        const float s = 1.0f / (1.0f + __expf(-u));
        partial[j] += s * we_r[t];
      }
    }
    // sum across the 16 lanes of each half (N dimension of the tile)
#pragma unroll
    for (int j = 0; j < 8; ++j) {
      float v = partial[j];
      v += __shfl_xor(v, 1, 32);
      v += __shfl_xor(v, 2, 32);
      v += __shfl_xor(v, 4, 32);
      v += __shfl_xor(v, 8, 32);
      partial[j] = v;
    }
    if (l16 == 0) {
      float* eo = e_out + (size_t)p * Mz + rowBase + 8 * half;
#pragma unroll
      for (int j = 0; j < 8; ++j) eo[j] = partial[j];
    }
  }
}

// ---------------- per-pair softmax stats: max and sum(exp(e - max)) ----------------
__global__ __launch_bounds__(256) void k_softmax_stats(const float* __restrict__ e,
                                                       float* __restrict__ stats) {
  const int p = blockIdx.x;
  const float* ep = e + (size_t)p * Mz;
  const int tid = threadIdx.x;
  __shared__ float redm[8];
  __shared__ float reds[8];

  float mx = -3.0e38f;
  for (int m = tid; m < Mz; m += 256) mx = fmaxf(mx, ep[m]);
#pragma unroll
  for (int off = 16; off > 0; off >>= 1) mx = fmaxf(mx, __shfl_xor(mx, off, 32));
  if ((tid & 31) == 0) redm[tid >> 5] = mx;
  __syncthreads();
  float mall = redm[0];
#pragma unroll
  for (int w = 1; w < 8; ++w) mall = fmaxf(mall, redm[w]);

  float s = 0.f;
  for (int m = tid; m < Mz; m += 256) s += __expf(ep[m] - mall);
#pragma unroll
  for (int off = 16; off > 0; off >>= 1) s += __shfl_xor(s, off, 32);
  if ((tid & 31) == 0) reds[tid >> 5] = s;
  __syncthreads();
  if (tid == 0) {
    float tot = 0.f;
#pragma unroll
    for (int w = 0; w < 8; ++w) tot += reds[w];
    stats[p * 2 + 0] = mall;
    stats[p * 2 + 1] = tot;
  }
}

// ---------------- partial weighted sums: part[p, chunk, d] = sum_{r in chunk} exp(e-mx)*x[b, r, d] ----------------
__global__ __launch_bounds__(128) void k_wsum_partial(const float* __restrict__ x,
                                                      const float* __restrict__ e,
                                                      const float* __restrict__ stats,
                                                      float* __restrict__ part) {
  const int p = blockIdx.y;
  const int b = p / Oz;
  const int chunk = blockIdx.x;       // 0..31, 256 rows each
  const int d = threadIdx.x;          // 0..127
  __shared__ float w[256];
  const float mall = stats[p * 2 + 0];
  const float* ep = e + (size_t)p * Mz + chunk * 256;
  w[d]       = __expf(ep[d]       - mall);
  w[d + 128] = __expf(ep[d + 128] - mall);
  __syncthreads();
  const float* xp = x + ((size_t)b * Mz + (size_t)chunk * 256) * Dz + d;
  float acc = 0.f;
#pragma unroll 8
  for (int r = 0; r < 256; ++r) acc += w[r] * xp[(size_t)r * Dz];
  part[((size_t)p * 32 + chunk) * Dz + d] = acc;
}

// ---------------- final reduction + 1/Z scale ----------------
__global__ __launch_bounds__(128) void k_wsum_final(const float* __restrict__ part,
                                                    const float* __restrict__ stats,
                                                    float* __restrict__ out) {
  const int p = blockIdx.x;
  const int d = threadIdx.x;
  const float invZ = 1.0f / stats[p * 2 + 1];
  float acc = 0.f;
#pragma unroll
  for (int c = 0; c < 32; ++c) acc += part[((size_t)p * 32 + c) * Dz + d];
  out[(size_t)p * Dz + d] = acc * invZ;
}

extern "C" void kernel_launch(void* const* d_in, const int* in_sizes, int n_in,
                              void* d_out, int out_size, void* d_ws, size_t ws_size,
                              hipStream_t stream) {
  const float* x    = (const float*)d_in[0];   // (B,O,N,D) f32
  const float* Wu   = (const float*)d_in[1];   // (O,D,D)
  const float* bu   = (const float*)d_in[2];   // (O,D)
  const float* Wv   = (const float*)d_in[3];   // (O,D,D)
  const float* We   = (const float*)d_in[4];   // (O,D)
  const int*   last = (const int*)d_in[5];     // (B,O)
  float* out = (float*)d_out;                  // (B,O,D) f32

  float* ws       = (float*)d_ws;
  float* ws_xv    = ws + OFF_XV;
  float* ws_e     = ws + OFF_E;
  float* ws_stats = ws + OFF_STATS;
  float* ws_part  = ws + OFF_PART;

  k_xv<<<dim3(PAIRS), dim3(Dz), 0, stream>>>(x, Wv, last, ws_xv);
  k_scores<<<dim3(8, PAIRS), dim3(256), 0, stream>>>(x, Wu, bu, We, ws_xv, ws_e);
  k_softmax_stats<<<dim3(PAIRS), dim3(256), 0, stream>>>(ws_e, ws_stats);
  k_wsum_partial<<<dim3(32, PAIRS), dim3(128), 0, stream>>>(x, ws_e, ws_stats, ws_part);
  k_wsum_final<<<dim3(PAIRS), dim3(128), 0, stream>>>(ws_part, ws_stats, out);
}